// Forward_Model_27616639714076
// MI455X (gfx1250) — compile-verified
//
#include <hip/hip_runtime.h>

typedef float v4f __attribute__((ext_vector_type(4)));

#define LDS_AS3 __attribute__((address_space(3)))

// out[e] = in[e] * q50[e % 64] + 1e-5, n = 51.2M f32 elements.
// Each thread: 1 q-chunk (float4, in regs) x 4 blocks of 64 -> 16 elements.
__global__ __launch_bounds__(256) void Forward_Model_27616639714076_kernel(
    const float* __restrict__ in, const float* __restrict__ q50,
    float* __restrict__ out, long long n)
{
    // 512B LDS: lanes 0..15 hold the real 256B q50 table; lanes 16..31 write
    // a duplicate into the upper half (keeps EXEC full, no OOB global reads).
    __shared__ __align__(16) float qs[128];

    // --- CDNA5 async tensor/LDS path: one b128 async copy per lane of wave 0.
    if (threadIdx.x < 32) {
        unsigned lane = threadIdx.x;
        unsigned lds_addr =
            (unsigned)(unsigned long long)(LDS_AS3 float*)&qs[0] + lane * 16u;
        const float* gaddr = q50 + (lane & 15u) * 4u;  // lanes 16..31 duplicate
        asm volatile(
            "global_load_async_to_lds_b128 %0, %1, off\n\t"
            "s_wait_asynccnt 0"
            :
            : "v"(lds_addr), "v"(gaddr)
            : "memory");
    }
    __syncthreads();

    long long t = (long long)blockIdx.x * blockDim.x + threadIdx.x;
    const int       l    = (int)(t & 15);       // which float4 of an 8x8 block
    const long long g    = t >> 4;              // group of 4 consecutive blocks
    const long long base = g * 256 + (long long)l * 4;

    // Per-thread quant chunk lives in 4 VGPRs for the whole kernel.
    v4f qv = *(const v4f*)&qs[l * 4];

    if (base + 195 < n) {
        // Fast path: 4 x b128 NT loads, 16 FMAs, 4 x b128 NT stores.
        const v4f* pin  = (const v4f*)(in  + base);
        v4f*       pout = (v4f*)      (out + base);
        v4f a0 = __builtin_nontemporal_load(pin +  0);   // block +0   (same q)
        v4f a1 = __builtin_nontemporal_load(pin + 16);   // block +64
        v4f a2 = __builtin_nontemporal_load(pin + 32);   // block +128
        v4f a3 = __builtin_nontemporal_load(pin + 48);   // block +192
        v4f r0 = a0 * qv + 1e-5f;
        v4f r1 = a1 * qv + 1e-5f;
        v4f r2 = a2 * qv + 1e-5f;
        v4f r3 = a3 * qv + 1e-5f;
        __builtin_nontemporal_store(r0, pout +  0);
        __builtin_nontemporal_store(r1, pout + 16);
        __builtin_nontemporal_store(r2, pout + 32);
        __builtin_nontemporal_store(r3, pout + 48);
    } else {
        // Tail (never taken for n = 51.2M, kept for arbitrary n).
        for (int k = 0; k < 4; ++k) {
            long long b = base + (long long)k * 64;
            for (int c = 0; c < 4; ++c) {
                long long e = b + c;
                if (e < n) out[e] = fmaf(in[e], qs[l * 4 + c], 1e-5f);
            }
        }
    }
}

extern "C" void kernel_launch(void* const* d_in, const int* in_sizes, int n_in,
                              void* d_out, int out_size, void* d_ws, size_t ws_size,
                              hipStream_t stream) {
    const float* in  = (const float*)d_in[0];   // [16*SIZE, 8, 8] f32
    const float* q50 = (const float*)d_in[1];   // [8, 8] f32
    float* out = (float*)d_out;                 // same element count as input

    long long n = (long long)in_sizes[0];       // 51,200,000
    long long threads = (n + 15) / 16;          // 16 elements per thread
    int block = 256;                            // 8 wave32s
    long long grid = (threads + block - 1) / block;  // 12,500 for SIZE=50000

    Forward_Model_27616639714076_kernel<<<(unsigned)grid, block, 0, stream>>>(
        in, q50, out, n);
}